// DeepBSDESR_25675314495639
// MI455X (gfx1250) — compile-verified
//
#include <hip/hip_runtime.h>
#include <math.h>

#define B_TOT   16384
#define NSTEPS  32
#define HID     128
#define DT_STEP (1.0f / 32.0f)

typedef __attribute__((ext_vector_type(2))) float v2f;
typedef __attribute__((ext_vector_type(8))) float v8f;

__device__ __forceinline__ v8f wmma4(v2f a, v2f b, v8f c) {
  // V_WMMA_F32_16X16X4_F32 : D(16x16 f32) = A(16x4) * B(4x16) + C
  return __builtin_amdgcn_wmma_f32_16x16x4_f32(
      /*neg_a=*/false, a, /*neg_b=*/false, b,
      /*c_mod=*/(short)0, c, /*reuse_a=*/false, /*reuse_b=*/false);
}

__device__ __forceinline__ float fast_tanh(float x) {
#if __has_builtin(__builtin_amdgcn_tanhf)
  return __builtin_amdgcn_tanhf(x);
#elif __has_builtin(__builtin_amdgcn_tanh_f32)
  return __builtin_amdgcn_tanh_f32(x);
#else
  return tanhf(x);
#endif
}

// Sum across the 16 lanes of each wave32 half (xor of lane bits 0..3 stays
// inside a half). After this every lane of a half holds the half's sum.
__device__ __forceinline__ float reduce_half16(float x) {
  x += __shfl_xor(x, 1, 32);
  x += __shfl_xor(x, 2, 32);
  x += __shfl_xor(x, 4, 32);
  x += __shfl_xor(x, 8, 32);
  return x;
}

// Branch-free 8-way select (cndmask tree); s[] indexed only by constants.
__device__ __forceinline__ float sel8(const float* s, int idx) {
  float a0 = (idx & 1) ? s[1] : s[0];
  float a1 = (idx & 1) ? s[3] : s[2];
  float a2 = (idx & 1) ? s[5] : s[4];
  float a3 = (idx & 1) ? s[7] : s[6];
  float b0 = (idx & 2) ? a1 : a0;
  float b1 = (idx & 2) ? a3 : a2;
  return (idx & 4) ? b1 : b0;
}

__global__ __launch_bounds__(256)
void bsde_scan_kernel(const float* __restrict__ pN,   // (33,B,1)
                      const float* __restrict__ pX,   // (33,B,3)
                      const float* __restrict__ tg,   // (33,)
                      const float* __restrict__ dBp,  // (32,B,3)
                      const float* __restrict__ rW1, const float* __restrict__ rb1,
                      const float* __restrict__ rW2, const float* __restrict__ rb2,
                      const float* __restrict__ gW1, const float* __restrict__ gb1,
                      const float* __restrict__ gW2, const float* __restrict__ gb2,
                      const float* __restrict__ jW1, const float* __restrict__ jb1,
                      const float* __restrict__ jW2, const float* __restrict__ jb2,
                      float* __restrict__ out)
{
  const int tid   = threadIdx.x;
  const int lane  = tid & 31;
  const int wave  = tid >> 5;
  const int pbase = (blockIdx.x * 8 + wave) * 16;

  const int  lm = lane & 15;       // A-row particle / B-col hidden index
  const bool loh = lane < 16;      // low half holds K=0,1 ; high half K=2,3
  const int  pA = pbase + lm;      // particle this lane feeds into A rows
  // particle owned by this lane for the per-particle scalar update:
  const int  q  = pbase + (lane & 7) + ((lane >> 1) & 8);
  const int  idx = lane & 7;
  const float kdelta = loh ? 1.0f : 0.0f;   // bump of feature 0 (lives in K=0)

  // ---------------- u0 = MLP_r(N0, X0) ----------------
  float u;
  {
    float a0 = loh ? pN[pA] : pX[pA * 3 + 1];
    float a1 = loh ? pX[pA * 3 + 0] : pX[pA * 3 + 2];
    v2f a = {a0, a1};
    float acc[8];
#pragma unroll
    for (int v = 0; v < 8; ++v) acc[v] = 0.0f;
#pragma unroll
    for (int t = 0; t < 8; ++t) {
      const float2 w = *(const float2*)(rW1 + (16 * t + lm) * 4 + (loh ? 0 : 2));
      v2f b = {w.x, w.y};
      v8f z = {};
      v8f d = wmma4(a, b, z);
      const float bias = rb1[16 * t + lm];
      const float w2   = rW2[16 * t + lm];
#pragma unroll
      for (int v = 0; v < 8; ++v) acc[v] += w2 * fast_tanh(d[v] + bias);
    }
#pragma unroll
    for (int v = 0; v < 8; ++v) acc[v] = reduce_half16(acc[v]);
    u = sel8(acc, idx) + rb2[0];
  }

  // ---------------- 32-step sequential scan ----------------
#pragma unroll 1
  for (int i = 0; i < NSTEPS; ++i) {
    const float* Ni = pN + (size_t)i * B_TOT;
    const float* Xi = pX + (size_t)i * B_TOT * 3;

    // A matrices: features (N, X0, X1, X2); N±1 only changes K=0 (low half v0)
    float a0 = loh ? Ni[pA] : Xi[pA * 3 + 1];
    float a1 = loh ? Xi[pA * 3 + 0] : Xi[pA * 3 + 2];
    v2f a  = {a0,          a1};
    v2f ap = {a0 + kdelta, a1};
    v2f am = {a0 - kdelta, a1};

    const float* gw1  = gW1 + (size_t)i * HID * 4;
    const float* gb1i = gb1 + (size_t)i * HID;
    const float* gw2  = gW2 + (size_t)i * 3 * HID;
    const float* jw1  = jW1 + (size_t)i * HID * 4;
    const float* jb1i = jb1 + (size_t)i * HID;
    const float* jw2  = jW2 + (size_t)i * HID;

    float pg0[8], pg1[8], pg2[8], pj[8], pjp[8], pjm[8];
#pragma unroll
    for (int v = 0; v < 8; ++v) {
      pg0[v] = pg1[v] = pg2[v] = 0.0f;
      pj[v] = pjp[v] = pjm[v] = 0.0f;
    }

#pragma unroll
    for (int t = 0; t < 8; ++t) {
      const int h = 16 * t + lm;
      const float2 wg = *(const float2*)(gw1 + h * 4 + (loh ? 0 : 2));
      const float2 wj = *(const float2*)(jw1 + h * 4 + (loh ? 0 : 2));
      v2f bg = {wg.x, wg.y};
      v2f bj = {wj.x, wj.y};
      v8f z = {};
      v8f dg  = wmma4(a,  bg, z);   // grad MLP hidden
      v8f dj  = wmma4(a,  bj, z);   // jump MLP at N
      v8f djp = wmma4(ap, bj, z);   // jump MLP at N+1
      v8f djm = wmma4(am, bj, z);   // jump MLP at N-1
      const float biasg = gb1i[h];
      const float biasj = jb1i[h];
      const float w20 = gw2[0 * HID + h];
      const float w21 = gw2[1 * HID + h];
      const float w22 = gw2[2 * HID + h];
      const float w2j = jw2[h];
#pragma unroll
      for (int v = 0; v < 8; ++v) {
        const float hg = fast_tanh(dg[v] + biasg);
        pg0[v] += w20 * hg;
        pg1[v] += w21 * hg;
        pg2[v] += w22 * hg;
        pj[v]  += w2j * fast_tanh(dj[v]  + biasj);
        pjp[v] += w2j * fast_tanh(djp[v] + biasj);
        pjm[v] += w2j * fast_tanh(djm[v] + biasj);
      }
    }

    // cross-lane sum over hidden columns, then pick owner particle's value
#pragma unroll
    for (int v = 0; v < 8; ++v) {
      pg0[v] = reduce_half16(pg0[v]);
      pg1[v] = reduce_half16(pg1[v]);
      pg2[v] = reduce_half16(pg2[v]);
      pj[v]  = reduce_half16(pj[v]);
      pjp[v] = reduce_half16(pjp[v]);
      pjm[v] = reduce_half16(pjm[v]);
    }
    const float gu0 = sel8(pg0, idx) + gb2[i * 3 + 0];
    const float gu1 = sel8(pg1, idx) + gb2[i * 3 + 1];
    const float gu2 = sel8(pg2, idx) + gb2[i * 3 + 2];
    const float jb  = jb2[i];
    const float ui  = sel8(pj,  idx) + jb;
    const float uip = sel8(pjp, idx) + jb;
    const float uim = sel8(pjm, idx) + jb;
    const float dp = uip - ui;
    const float dm = uim - ui;

    // -------- per-particle epilogue (branch-free; EXEC stays full) --------
    const float Nq  = Ni[q];
    const float x0  = Xi[q * 3 + 0];
    const float x1  = Xi[q * 3 + 1];
    const float x2  = Xi[q * 3 + 2];
    const float* dBq = dBp + ((size_t)i * B_TOT + q) * 3;
    const float db0 = dBq[0], db1 = dBq[1], db2v = dBq[2];
    const float Np1 = pN[(size_t)(i + 1) * B_TOT + q];
    const float ti  = tg[i];

    const float alpha = 0.3f / (1.0f + __expf(-0.1f * Nq));
    const float beta  = 0.2f / (1.0f + __expf( 0.1f * Nq));
    const float kk = __fsqrt_rn(2.0f / (1.0f + 0.1f * Nq * Nq));
    // t1 = (I - x x^T) dB ; t2 = (I + x x^T) t1 ; grad_bmm = k*grad_u . t2
    const float xdb = x0 * db0 + x1 * db1 + x2 * db2v;
    const float t10 = db0 - x0 * xdb, t11 = db1 - x1 * xdb, t12 = db2v - x2 * xdb;
    const float xt1 = x0 * t10 + x1 * t11 + x2 * t12;
    const float t20 = t10 + x0 * xt1, t21 = t11 + x1 * xt1, t22 = t12 + x2 * xt1;
    const float grad_bmm = kk * (gu0 * t20 + gu1 * t21 + gu2 * t22);

    const float f = -0.05f * u + 0.05f * (gu0 * x0 + gu1 * x1 + gu2 * x2)
                    + 0.01f * ti;
    u = u + grad_bmm - (alpha * dp + beta * dm) * DT_STEP - f * DT_STEP;

    const float dN = rintf(Np1 - Nq);                 // jnp.round
    u += (dN > 0.0f ? dp : 0.0f) + (dN < 0.0f ? dm : 0.0f);
  }

  // ---------------- terminal g and store ----------------
  const float* Nl = pN + (size_t)NSTEPS * B_TOT;
  const float* Xl = pX + (size_t)NSTEPS * B_TOT * 3;
  const float gout = 0.1f * Nl[q] + cosf(Xl[q * 3 + 0]);

  if ((lane & 8) == 0) {     // one owner lane per particle
    out[q]         = u;      // output 0: u  (B,1)
    out[B_TOT + q] = gout;   // output 1: g  (B,1)
  }
}

extern "C" void kernel_launch(void* const* d_in, const int* in_sizes, int n_in,
                              void* d_out, int out_size, void* d_ws, size_t ws_size,
                              hipStream_t stream) {
  const float* pN  = (const float*)d_in[0];
  const float* pX  = (const float*)d_in[1];
  const float* tg  = (const float*)d_in[2];
  const float* dBp = (const float*)d_in[3];
  const float* rW1 = (const float*)d_in[4];
  const float* rb1 = (const float*)d_in[5];
  const float* rW2 = (const float*)d_in[6];
  const float* rb2 = (const float*)d_in[7];
  const float* gW1 = (const float*)d_in[8];
  const float* gb1 = (const float*)d_in[9];
  const float* gW2 = (const float*)d_in[10];
  const float* gb2 = (const float*)d_in[11];
  const float* jW1 = (const float*)d_in[12];
  const float* jb1 = (const float*)d_in[13];
  const float* jW2 = (const float*)d_in[14];
  const float* jb2 = (const float*)d_in[15];

  dim3 block(256);                 // 8 waves -> 128 particles per block
  dim3 grid(B_TOT / 128);          // 128 blocks
  bsde_scan_kernel<<<grid, block, 0, stream>>>(
      pN, pX, tg, dBp, rW1, rb1, rW2, rb2,
      gW1, gb1, gW2, gb2, jW1, jb1, jW2, jb2,
      (float*)d_out);
}